// Dummy_Net_36515811950832
// MI455X (gfx1250) — compile-verified
//
#include <hip/hip_runtime.h>
#include <math.h>

typedef float v2f __attribute__((ext_vector_type(2)));
typedef float v8f __attribute__((ext_vector_type(8)));

#define NNODES 20000
#define NEDGES 250000

// ---------------------------------------------------------------- utilities

__device__ __forceinline__ float atomicMaxF(float* addr, float val) {
  // IEEE-order trick: int max for >=0, uint min for <0. Init must be -inf.
  if (val >= 0.0f)
    return __int_as_float(atomicMax((int*)addr, __float_as_int(val)));
  else
    return __uint_as_float(atomicMin((unsigned int*)addr, __float_as_uint(val)));
}

__global__ void fill_kernel(float* __restrict__ p, float v, int n) {
  int i = blockIdx.x * blockDim.x + threadIdx.x;
  if (i < n) p[i] = v;
}

// ------------------------------------------------------- CDNA5 async helpers

// Async global->LDS 16B copy (GLOBAL_LOAD_ASYNC_TO_LDS_B128, ASYNCcnt).
// ldsOff: LDS byte offset (low 32 bits of a generic LDS pointer per ISA 10.2).
__device__ __forceinline__ void async_copy_b128(unsigned ldsOff,
                                                const float* gptr) {
  unsigned long long ga = (unsigned long long)gptr;
  asm volatile("global_load_async_to_lds_b128 %0, %1, off"
               :: "v"(ldsOff), "v"(ga) : "memory");
}

__device__ __forceinline__ void wait_async_zero() {
  asm volatile("s_wait_asynccnt 0x0" ::: "memory");
}

// ---------------------------------------------------- fp32 WMMA GEMM kernels
//
// fp32 WMMA 16x16x4 layouts (ISA 7.12.2):
//   A (16x4):  lane l holds row (l&15), K = (l>>4)*2 + vgpr   -> v2f
//   B (4x16):  lane l holds col (l&15), K = vgpr + (l>>4)*2   -> v2f
//   C (16x16): vgpr i: lanes 0-15 row i, lanes 16-31 row i+8  -> v8f

// Generic: one wave computes one 16x16 tile of C = A[M,K]*B[K,NC] + bias.
__global__ __launch_bounds__(128)
void gemm_wmma(const float* __restrict__ A, const float* __restrict__ B,
               const float* __restrict__ bias, float* __restrict__ C,
               int M, int K, int NC) {
  const int ntiles = NC >> 4;
  const int total = (M >> 4) * ntiles;
  int wave = blockIdx.x * (blockDim.x >> 5) + (threadIdx.x >> 5);
  if (wave >= total) return;                 // wave-uniform, EXEC stays full
  int mt = wave / ntiles, nt = wave % ntiles;

  int lane = threadIdx.x & 31;
  int hi = lane >> 4;                        // 0: K pair {0,1}; 1: {2,3}
  int lm = lane & 15;

  const float* arow = A + (size_t)(mt * 16 + lm) * K + hi * 2;
  const float* bcol = B + nt * 16 + lm;

  v8f acc = {};
  for (int k = 0; k < K; k += 4) {
    v2f a, b;
    a.x = arow[k];
    a.y = arow[k + 1];
    b.x = bcol[(size_t)(k + hi * 2) * NC];
    b.y = bcol[(size_t)(k + hi * 2 + 1) * NC];
    acc = __builtin_amdgcn_wmma_f32_16x16x4_f32(false, a, false, b,
                                                (short)0, acc, false, false);
  }
  float bv = bias ? bias[nt * 16 + lm] : 0.0f;
  int col = nt * 16 + lm;
  int rbase = mt * 16 + hi * 8;
#pragma unroll
  for (int i = 0; i < 8; ++i)
    C[(size_t)(rbase + i) * NC + col] = acc[i] + bv;
}

// conv1 projections: block = one 16-row M tile of one matrix (blockIdx.y
// selects Wq/Wk/Wv/Wskip). The 16x128 A tile is async-copied to LDS once and
// shared by 8 waves; each wave computes 4 N-tiles (wave's 4 WMMAs per K-step
// share one A fragment read from LDS).
__global__ __launch_bounds__(256)
void proj4_wmma(const float* __restrict__ x,
                const float* W0, const float* W1, const float* W2, const float* W3,
                const float* b0, const float* b1, const float* b2, const float* b3,
                float* o0, float* o1, float* o2, float* o3) {
  const int K = 128, NC = 512;
  __shared__ float tileA[16 * 128];          // 8 KB

  int mt = blockIdx.x;                       // 0..1249
  const float *W, *bias; float* out;
  switch (blockIdx.y) {
    case 0:  W = W0; bias = b0; out = o0; break;
    case 1:  W = W1; bias = b1; out = o1; break;
    case 2:  W = W2; bias = b2; out = o2; break;
    default: W = W3; bias = b3; out = o3; break;
  }

  // stage A tile (2048 floats = 512 x b128) via async loads to LDS
  {
    const float* gbase = x + (size_t)mt * 16 * K;
    for (int c = threadIdx.x; c < 512; c += 256) {
      unsigned ldsOff = (unsigned)(size_t)(&tileA[c * 4]);
      async_copy_b128(ldsOff, gbase + c * 4);
    }
  }
  wait_async_zero();
  __syncthreads();

  int wave = threadIdx.x >> 5;               // 0..7 -> N-tiles wave*4..wave*4+3
  int lane = threadIdx.x & 31;
  int hi = lane >> 4, lm = lane & 15;
  int ntBase = wave * 4;

  const float* arow = &tileA[lm * K + hi * 2];

  v8f acc0 = {}, acc1 = {}, acc2 = {}, acc3 = {};
  for (int k = 0; k < K; k += 4) {
    v2f a = *(const v2f*)&arow[k];           // ds_load_b64 (8B aligned)
    const float* br0 = W + (size_t)(k + hi * 2) * NC + lm;
    const float* br1 = br0 + NC;
    v2f bb0, bb1, bb2, bb3;
    bb0.x = br0[(ntBase + 0) * 16]; bb0.y = br1[(ntBase + 0) * 16];
    bb1.x = br0[(ntBase + 1) * 16]; bb1.y = br1[(ntBase + 1) * 16];
    bb2.x = br0[(ntBase + 2) * 16]; bb2.y = br1[(ntBase + 2) * 16];
    bb3.x = br0[(ntBase + 3) * 16]; bb3.y = br1[(ntBase + 3) * 16];
    acc0 = __builtin_amdgcn_wmma_f32_16x16x4_f32(false, a, false, bb0,
                                                 (short)0, acc0, false, false);
    acc1 = __builtin_amdgcn_wmma_f32_16x16x4_f32(false, a, false, bb1,
                                                 (short)0, acc1, false, false);
    acc2 = __builtin_amdgcn_wmma_f32_16x16x4_f32(false, a, false, bb2,
                                                 (short)0, acc2, false, false);
    acc3 = __builtin_amdgcn_wmma_f32_16x16x4_f32(false, a, false, bb3,
                                                 (short)0, acc3, false, false);
  }

  int rbase = mt * 16 + hi * 8;
#pragma unroll
  for (int tIdx = 0; tIdx < 4; ++tIdx) {
    v8f acc = (tIdx == 0) ? acc0 : (tIdx == 1) ? acc1 : (tIdx == 2) ? acc2 : acc3;
    int col = (ntBase + tIdx) * 16 + lm;
    float bv = bias[col];
#pragma unroll
    for (int i = 0; i < 8; ++i)
      out[(size_t)(rbase + i) * NC + col] = acc[i] + bv;
  }
}

// ----------------------------------------------------- conv1 edge kernels

// one wave per edge: per-head dot(q[dst], k[src]+ea*We) -> alpha, atomic max
__global__ __launch_bounds__(128)
void edge_alpha1(const int* __restrict__ ei, const float* __restrict__ ea,
                 const float* __restrict__ Q, const float* __restrict__ Kf,
                 const float* __restrict__ We, float* __restrict__ alpha,
                 float* __restrict__ amax) {
  int e = blockIdx.x * (blockDim.x >> 5) + (threadIdx.x >> 5);
  if (e >= NEDGES) return;
  int lane = threadIdx.x & 31;
  int src = ei[e], dst = ei[NEDGES + e];
  float eav = ea[e];
  const float* q = Q + (size_t)dst * 512;
  const float* k = Kf + (size_t)src * 512;
  int base = lane * 16;                 // lanes 0-7: head0, 8-15: head1, ...
  float s = 0.0f;
#pragma unroll
  for (int j = 0; j < 16; ++j) {
    int ch = base + j;
    s += q[ch] * (k[ch] + eav * We[ch]);
  }
  s += __shfl_xor(s, 1, 32);
  s += __shfl_xor(s, 2, 32);
  s += __shfl_xor(s, 4, 32);
  if ((lane & 7) == 0) {
    int h = lane >> 3;
    float a = s * 0.08838834764831845f;  // 1/sqrt(128)
    alpha[(size_t)e * 4 + h] = a;
    atomicMaxF(&amax[(size_t)dst * 4 + h], a);
  }
}

// one thread per (edge,head): exp(alpha - amax[dst]) and denominator sum
__global__ void edge_exp(const int* __restrict__ ei, float* __restrict__ alpha,
                         const float* __restrict__ amax, float* __restrict__ den) {
  int i = blockIdx.x * blockDim.x + threadIdx.x;
  if (i >= NEDGES * 4) return;
  int e = i >> 2, h = i & 3;
  int dst = ei[NEDGES + e];
  float v = __expf(alpha[i] - amax[(size_t)dst * 4 + h]);
  alpha[i] = v;
  atomicAdd(&den[(size_t)dst * 4 + h], v);
}

// one wave per edge: scatter (v[src]+e)*alpha into out[dst]
__global__ __launch_bounds__(128)
void edge_scatter1(const int* __restrict__ ei, const float* __restrict__ ea,
                   const float* __restrict__ V, const float* __restrict__ We,
                   const float* __restrict__ ealpha, const float* __restrict__ den,
                   float* __restrict__ out) {
  int e = blockIdx.x * (blockDim.x >> 5) + (threadIdx.x >> 5);
  if (e >= NEDGES) return;
  int lane = threadIdx.x & 31;
  int src = ei[e], dst = ei[NEDGES + e];
  float eav = ea[e];
  int h = lane >> 3;
  float w = ealpha[(size_t)e * 4 + h] / (den[(size_t)dst * 4 + h] + 1e-16f);
  const float* v = V + (size_t)src * 512;
  float* o = out + (size_t)dst * 512;
  int base = lane * 16;
#pragma unroll
  for (int j = 0; j < 16; ++j) {
    int ch = base + j;
    atomicAdd(&o[ch], (v[ch] + eav * We[ch]) * w);
  }
}

// one wave per node: beta = sigmoid([out,skip,out-skip] . Wbeta); gate
__global__ __launch_bounds__(128)
void beta_gate1(const float* __restrict__ out, const float* __restrict__ skip,
                const float* __restrict__ Wb, float* __restrict__ hout) {
  int n = blockIdx.x * (blockDim.x >> 5) + (threadIdx.x >> 5);
  if (n >= NNODES) return;
  int lane = threadIdx.x & 31;
  const float* o = out + (size_t)n * 512;
  const float* sk = skip + (size_t)n * 512;
  int base = lane * 16;
  float s = 0.0f;
#pragma unroll
  for (int j = 0; j < 16; ++j) {
    int ch = base + j;
    float ov = o[ch], sv = sk[ch];
    s += ov * Wb[ch] + sv * Wb[512 + ch] + (ov - sv) * Wb[1024 + ch];
  }
  for (int m = 1; m < 32; m <<= 1) s += __shfl_xor(s, m, 32);
  float beta = 1.0f / (1.0f + __expf(-s));
  float* ho = hout + (size_t)n * 512;
#pragma unroll
  for (int j = 0; j < 16; ++j) {
    int ch = base + j;
    ho[ch] = beta * sk[ch] + (1.0f - beta) * o[ch];
  }
}

// ---------------------------------------------------------------- batchnorm

__global__ void bn_stats(const float* __restrict__ h, float* __restrict__ st,
                         int M, int C) {
  int c = threadIdx.x;                        // blockDim.x == C
  float s = 0.0f, sq = 0.0f;
  for (int r = blockIdx.x; r < M; r += gridDim.x) {
    float v = h[(size_t)r * C + c];
    s += v; sq += v * v;
  }
  atomicAdd(&st[c], s);
  atomicAdd(&st[C + c], sq);
}

__global__ void bn_apply(float* __restrict__ h, const float* __restrict__ st,
                         const float* __restrict__ g, const float* __restrict__ b,
                         int M, int C) {
  int i = blockIdx.x * blockDim.x + threadIdx.x;
  if (i >= M * C) return;
  int c = i % C;
  float m = st[c] / (float)M;
  float var = st[C + c] / (float)M - m * m;
  h[i] = (h[i] - m) * rsqrtf(var + 1e-5f) * g[c] + b[c];
}

// ------------------------------------------------------------- conv2 (hc=4)

// pack the four [128,4] weights into one [128,16] matrix (+ packed bias)
__global__ void pack_conv2(const float* Wq, const float* Wk, const float* Wv,
                           const float* Ws, const float* bq, const float* bk,
                           const float* bv, const float* bs,
                           float* __restrict__ Wcat, float* __restrict__ bcat) {
  int i = blockIdx.x * blockDim.x + threadIdx.x;
  if (i < 128 * 16) {
    int k = i >> 4, c = i & 15;
    int m = c >> 2, lc = c & 3;
    const float* W = (m == 0) ? Wq : (m == 1) ? Wk : (m == 2) ? Wv : Ws;
    Wcat[k * 16 + c] = W[k * 4 + lc];
  }
  if (i < 16) {
    int m = i >> 2, lc = i & 3;
    const float* b = (m == 0) ? bq : (m == 1) ? bk : (m == 2) ? bv : bs;
    bcat[i] = b[lc];
  }
}

// P layout per node: [0..3]=q2, [4..7]=k2, [8..11]=v2, [12..15]=skip2
__global__ void edge_alpha2(const int* __restrict__ ei, const float* __restrict__ ea,
                            const float* __restrict__ P, const float* __restrict__ We2,
                            float* __restrict__ alpha, float* __restrict__ amax) {
  int e = blockIdx.x * blockDim.x + threadIdx.x;
  if (e >= NEDGES) return;
  int src = ei[e], dst = ei[NEDGES + e];
  float eav = ea[e];
#pragma unroll
  for (int h = 0; h < 4; ++h) {
    float a = P[(size_t)dst * 16 + h] * (P[(size_t)src * 16 + 4 + h] + eav * We2[h]);
    alpha[(size_t)e * 4 + h] = a;    // cout==1 -> scale 1
    atomicMaxF(&amax[(size_t)dst * 4 + h], a);
  }
}

__global__ void edge_scatter2(const int* __restrict__ ei, const float* __restrict__ ea,
                              const float* __restrict__ P, const float* __restrict__ We2,
                              const float* __restrict__ ealpha, const float* __restrict__ den,
                              float* __restrict__ out) {
  int e = blockIdx.x * blockDim.x + threadIdx.x;
  if (e >= NEDGES) return;
  int src = ei[e], dst = ei[NEDGES + e];
  float eav = ea[e];
#pragma unroll
  for (int h = 0; h < 4; ++h) {
    float w = ealpha[(size_t)e * 4 + h] / (den[(size_t)dst * 4 + h] + 1e-16f);
    atomicAdd(&out[(size_t)dst * 4 + h],
              (P[(size_t)src * 16 + 8 + h] + eav * We2[h]) * w);
  }
}

// beta gate (layer 2) fused with transf2 linear -> pre-BN scalar per node
__global__ void beta2_transf2(const float* __restrict__ out2, const float* __restrict__ P,
                              const float* __restrict__ Wb2, const float* __restrict__ W2,
                              const float* __restrict__ b2, float* __restrict__ pre) {
  int n = blockIdx.x * blockDim.x + threadIdx.x;
  if (n >= NNODES) return;
  float s = 0.0f;
#pragma unroll
  for (int h = 0; h < 4; ++h) {
    float o = out2[(size_t)n * 4 + h], sk = P[(size_t)n * 16 + 12 + h];
    s += o * Wb2[h] + sk * Wb2[4 + h] + (o - sk) * Wb2[8 + h];
  }
  float beta = 1.0f / (1.0f + __expf(-s));
  float y = b2[0];
#pragma unroll
  for (int h = 0; h < 4; ++h) {
    float o = out2[(size_t)n * 4 + h], sk = P[(size_t)n * 16 + 12 + h];
    y += (beta * sk + (1.0f - beta) * o) * W2[h];
  }
  pre[n] = y;
}

__global__ void bn2_stats(const float* __restrict__ pre, float* __restrict__ st) {
  __shared__ float ss[256], sq[256];
  int t = threadIdx.x;
  float s = 0.0f, q = 0.0f;
  for (int i = blockIdx.x * 256 + t; i < NNODES; i += gridDim.x * 256) {
    float v = pre[i]; s += v; q += v * v;
  }
  ss[t] = s; sq[t] = q; __syncthreads();
  for (int k = 128; k > 0; k >>= 1) {
    if (t < k) { ss[t] += ss[t + k]; sq[t] += sq[t + k]; }
    __syncthreads();
  }
  if (t == 0) { atomicAdd(&st[0], ss[0]); atomicAdd(&st[1], sq[0]); }
}

__global__ void bn2_apply(const float* __restrict__ pre, const float* __restrict__ st,
                          const float* __restrict__ g, const float* __restrict__ b,
                          float* __restrict__ out) {
  int n = blockIdx.x * blockDim.x + threadIdx.x;
  if (n >= NNODES) return;
  float m = st[0] / (float)NNODES;
  float var = st[1] / (float)NNODES - m * m;
  out[n] = (pre[n] - m) * rsqrtf(var + 1e-5f) * g[0] + b[0];
}

// ------------------------------------------------------------------ launch

extern "C" void kernel_launch(void* const* d_in, const int* in_sizes, int n_in,
                              void* d_out, int out_size, void* d_ws, size_t ws_size,
                              hipStream_t stream) {
  const int N = NNODES, E = NEDGES;
  const float* x   = (const float*)d_in[0];
  const int*   ei  = (const int*)d_in[1];           // [0..E)=src, [E..2E)=dst
  const float* ea  = (const float*)d_in[2];
  // conv1
  const float* Wq1 = (const float*)d_in[3];  const float* bq1 = (const float*)d_in[4];
  const float* Wk1 = (const float*)d_in[5];  const float* bk1 = (const float*)d_in[6];
  const float* Wv1 = (const float*)d_in[7];  const float* bv1 = (const float*)d_in[8];
  const float* We1 = (const float*)d_in[9];
  const float* Ws1 = (const float*)d_in[10]; const float* bs1 = (const float*)d_in[11];
  const float* Wb1 = (const float*)d_in[12];
  // transf1 / bn1
  const float* Wt1 = (const float*)d_in[13]; const float* bt1 = (const float*)d_in[14];
  const float* g1  = (const float*)d_in[15]; const float* be1 = (const float*)d_in[16];
  // conv2
  const float* Wq2 = (const float*)d_in[17]; const float* bq2 = (const float*)d_in[18];
  const float* Wk2 = (const float*)d_in[19]; const float* bk2 = (const float*)d_in[20];
  const float* Wv2 = (const float*)d_in[21]; const float* bv2 = (const float*)d_in[22];
  const float* We2 = (const float*)d_in[23];
  const float* Ws2 = (const float*)d_in[24]; const float* bs2 = (const float*)d_in[25];
  const float* Wb2 = (const float*)d_in[26];
  // transf2 / bn2
  const float* Wt2 = (const float*)d_in[27]; const float* bt2 = (const float*)d_in[28];
  const float* g2  = (const float*)d_in[29]; const float* be2 = (const float*)d_in[30];

  float* ws = (float*)d_ws;
  size_t off = 0;
  auto alloc = [&](size_t n) { float* p = ws + off; off += n; return p; };
  float* Q     = alloc((size_t)N * 512);
  float* Kb    = alloc((size_t)N * 512);
  float* Vb    = alloc((size_t)N * 512);
  float* Sb    = alloc((size_t)N * 512);
  float* OUT1  = alloc((size_t)N * 512);
  float* AMAX1 = alloc((size_t)N * 4);
  float* DEN1  = alloc((size_t)N * 4);
  float* EAL1  = alloc((size_t)E * 4);
  float* H2    = alloc((size_t)N * 128);
  float* ST1   = alloc(256);
  float* W2CAT = alloc(128 * 16);
  float* B2CAT = alloc(16);
  float* P2    = alloc((size_t)N * 16);
  float* OUT2  = alloc((size_t)N * 4);
  float* AMAX2 = alloc((size_t)N * 4);
  float* DEN2  = alloc((size_t)N * 4);
  float* EAL2  = alloc((size_t)E * 4);
  float* PRE   = alloc((size_t)N);
  float* ST2   = alloc(8);
  (void)ws_size; (void)n_in; (void)in_sizes; (void)out_size;

  const float NEGINF = -__builtin_huge_valf();
  auto cdiv = [](int a, int b) { return (a + b - 1) / b; };

  // deterministic re-init of accumulators each call
  fill_kernel<<<cdiv(N * 512, 256), 256, 0, stream>>>(OUT1, 0.0f, N * 512);
  fill_kernel<<<cdiv(N * 4, 256), 256, 0, stream>>>(AMAX1, NEGINF, N * 4);
  fill_kernel<<<cdiv(N * 4, 256), 256, 0, stream>>>(DEN1, 0.0f, N * 4);
  fill_kernel<<<cdiv(N * 4, 256), 256, 0, stream>>>(OUT2, 0.0f, N * 4);
  fill_kernel<<<cdiv(N * 4, 256), 256, 0, stream>>>(AMAX2, NEGINF, N * 4);
  fill_kernel<<<cdiv(N * 4, 256), 256, 0, stream>>>(DEN2, 0.0f, N * 4);
  fill_kernel<<<1, 256, 0, stream>>>(ST1, 0.0f, 256);
  fill_kernel<<<1, 8, 0, stream>>>(ST2, 0.0f, 8);

  // conv1 projections: Q/K/V/SKIP (fp32 WMMA; A tile staged via async->LDS)
  proj4_wmma<<<dim3(N / 16, 4), 256, 0, stream>>>(
      x, Wq1, Wk1, Wv1, Ws1, bq1, bk1, bv1, bs1, Q, Kb, Vb, Sb);

  // conv1 attention
  edge_alpha1<<<cdiv(E, 4), 128, 0, stream>>>(ei, ea, Q, Kb, We1, EAL1, AMAX1);
  edge_exp<<<cdiv(E * 4, 256), 256, 0, stream>>>(ei, EAL1, AMAX1, DEN1);
  edge_scatter1<<<cdiv(E, 4), 128, 0, stream>>>(ei, ea, Vb, We1, EAL1, DEN1, OUT1);
  beta_gate1<<<cdiv(N, 4), 128, 0, stream>>>(OUT1, Sb, Wb1, Q);  // gated -> Q (reused)

  // transf1 + bn1
  {
    int waves = (N / 16) * (128 / 16);
    gemm_wmma<<<cdiv(waves, 4), 128, 0, stream>>>(Q, Wt1, bt1, H2, N, 512, 128);
  }
  bn_stats<<<256, 128, 0, stream>>>(H2, ST1, N, 128);
  bn_apply<<<cdiv(N * 128, 256), 256, 0, stream>>>(H2, ST1, g1, be1, N, 128);

  // conv2 projections (packed 128x16 GEMM, fp32 WMMA)
  pack_conv2<<<cdiv(128 * 16, 256), 256, 0, stream>>>(
      Wq2, Wk2, Wv2, Ws2, bq2, bk2, bv2, bs2, W2CAT, B2CAT);
  {
    int waves = (N / 16) * (16 / 16);
    gemm_wmma<<<cdiv(waves, 4), 128, 0, stream>>>(H2, W2CAT, B2CAT, P2, N, 128, 16);
  }
  // conv2 attention
  edge_alpha2<<<cdiv(E, 256), 256, 0, stream>>>(ei, ea, P2, We2, EAL2, AMAX2);
  edge_exp<<<cdiv(E * 4, 256), 256, 0, stream>>>(ei, EAL2, AMAX2, DEN2);
  edge_scatter2<<<cdiv(E, 256), 256, 0, stream>>>(ei, ea, P2, We2, EAL2, DEN2, OUT2);

  // beta gate 2 + transf2 + bn2 -> output [N,1]
  beta2_transf2<<<cdiv(N, 256), 256, 0, stream>>>(OUT2, P2, Wb2, Wt2, bt2, PRE);
  bn2_stats<<<128, 256, 0, stream>>>(PRE, ST2);
  bn2_apply<<<cdiv(N, 256), 256, 0, stream>>>(PRE, ST2, g2, be2, (float*)d_out);
}